// ROCKET_44023414784547
// MI455X (gfx1250) — compile-verified
//
#include <hip/hip_runtime.h>
#include <stdint.h>

// ================= problem constants (from reference) =================
#define N_KERNELS 10000
#define SEQ_LEN   512
#define KMAX      11
#define BATCH     16
// group key = (pad<<9)|(ksidx<<7)|dil : pad in {0,1}, ksidx in {0,1,2}, dil<=85<128.
// (pad, ks, dil) fully determines p2 = pad*dil*(ks-1) and L = 512+4p-dil*(ks-1),
// so every row of a tile shares dil, p2 AND L (uniform validity masking).
#define NKEYS     1024
#define MAX_TILES 1536           // bound: <=510 possible keys + 10000/16 tiles

typedef __attribute__((ext_vector_type(2))) float v2f;
typedef __attribute__((ext_vector_type(8))) float v8f;

// ================= workspace layout (int32 units, ~235 KB) ============
#define OFF_META 0                          // [0] = numTiles
#define OFF_KS   16
#define OFF_DIL  (OFF_KS  + N_KERNELS)
#define OFF_P2   (OFF_DIL + N_KERNELS)      // stores 2*p
#define OFF_L    (OFF_P2  + N_KERNELS)
#define OFF_SORT (OFF_L   + N_KERNELS)
#define OFF_HIST (OFF_SORT+ N_KERNELS)
#define OFF_TILE (OFF_HIST+ NKEYS)          // 5 ints per tile: base,rows,dil,p2,L

// ============ NumPy default_rng(0) == SeedSequence(0) + PCG64 =========
typedef unsigned __int128 u128;

struct Pcg64 { u128 state, inc; uint32_t buf; int has32; };

__device__ inline u128 pcg_mult() {
  return (((u128)2549297995355413924ULL) << 64) | (u128)4865540595714422341ULL;
}
__device__ inline uint64_t pcg_next64(Pcg64 &g) {
  g.state = g.state * pcg_mult() + g.inc;
  uint64_t hi = (uint64_t)(g.state >> 64), lo = (uint64_t)g.state;
  unsigned rot = (unsigned)(g.state >> 122);
  uint64_t x = hi ^ lo;
  return (x >> rot) | (x << ((64u - rot) & 63u));
}
__device__ inline uint32_t pcg_next32(Pcg64 &g) {   // numpy's has_uint32 buffering
  if (g.has32) { g.has32 = 0; return g.buf; }
  uint64_t v = pcg_next64(g);
  g.has32 = 1; g.buf = (uint32_t)(v >> 32);
  return (uint32_t)v;
}
__device__ inline double pcg_double(Pcg64 &g) {
  return (double)(pcg_next64(g) >> 11) * (1.0 / 9007199254740992.0);
}
// numpy SeedSequence hash constants
__device__ inline uint32_t ss_hashmix(uint32_t v, uint32_t &hc) {
  v ^= hc; hc *= 0x931e8875u; v *= hc; v ^= v >> 16; return v;
}
__device__ inline uint32_t ss_mix(uint32_t x, uint32_t y) {
  uint32_t r = x * 0xca01f9ddu; r ^= y * 0x4973f715u; r ^= r >> 16; return r;
}
__device__ inline void pcg_seed_from_seedseq0(Pcg64 &g) {
  uint32_t pool[4];
  uint32_t hc = 0x43b0d7e5u;                       // INIT_A
  for (int i = 0; i < 4; ++i) pool[i] = ss_hashmix(0u, hc);   // entropy = [0]
  for (int s = 0; s < 4; ++s)
    for (int d = 0; d < 4; ++d)
      if (s != d) pool[d] = ss_mix(pool[d], ss_hashmix(pool[s], hc));
  uint32_t w[8]; uint32_t hb = 0x8b51f9ddu;        // INIT_B
  for (int i = 0; i < 8; ++i) {
    uint32_t dv = pool[i & 3];
    dv ^= hb; hb *= 0x58f38dedu; dv *= hb; dv ^= dv >> 16;
    w[i] = dv;
  }
  uint64_t s64[4];
  for (int i = 0; i < 4; ++i) s64[i] = (uint64_t)w[2*i] | ((uint64_t)w[2*i+1] << 32);
  u128 initstate = ((u128)s64[0] << 64) | s64[1];
  u128 initseq   = ((u128)s64[2] << 64) | s64[3];
  g.state = 0; g.inc = (initseq << 1) | 1;
  g.state = g.state * pcg_mult() + g.inc;          // step
  g.state += initstate;
  g.state = g.state * pcg_mult() + g.inc;          // step
  g.has32 = 0; g.buf = 0;
}
// Lemire bounded draw on next_uint32, numpy Generator path (use_masked=False)
__device__ inline uint32_t lemire32(Pcg64 &g, uint32_t rng) {  // uniform in [0,rng]
  uint32_t re = rng + 1u;
  uint64_t m = (uint64_t)pcg_next32(g) * (uint64_t)re;
  uint32_t lo = (uint32_t)m;
  if (lo < re) {
    uint32_t thr = (0xFFFFFFFFu - rng) % re;
    while (lo < thr) { m = (uint64_t)pcg_next32(g) * (uint64_t)re; lo = (uint32_t)m; }
  }
  return (uint32_t)(m >> 32);
}

__device__ inline int group_key(const int* ws, int q) {
  int pad   = ws[OFF_P2 + q] > 0;
  int ksidx = (ws[OFF_KS + q] - 7) >> 1;           // 0,1,2
  return (pad << 9) | (ksidx << 7) | ws[OFF_DIL + q];
}

// =========== init: regenerate params, bin kernels, build WMMA tiles ===========
__global__ void rocket_init(int* __restrict__ ws) {
  const int tid = threadIdx.x;
  for (int i = tid; i < NKEYS; i += blockDim.x) ws[OFF_HIST + i] = 0;
  __syncthreads();
  if (tid != 0) return;

  Pcg64 g; pcg_seed_from_seedseq0(g);
  // 1) ks = rng.choice([7,9,11], 10000)  -> integers(0,3) via Lemire32
  for (int q = 0; q < N_KERNELS; ++q)
    ws[OFF_KS + q] = 7 + 2 * (int)lemire32(g, 2u);
  // 2) dil = floor(2 ** uniform(0, log2(511 // (ks-1))))
  for (int q = 0; q < N_KERNELS; ++q) {
    int ksv = ws[OFF_KS + q];
    double hi = log2((double)((SEQ_LEN - 1) / (ksv - 1)));
    double u = pcg_double(g);
    ws[OFF_DIL + q] = (int)floor(pow(2.0, hi * u));
  }
  // 3) use_pad = integers(0, 2);  p = (ks-1)*dil//2 * pad;  L = 512 + 4p - dil*(ks-1)
  for (int q = 0; q < N_KERNELS; ++q) {
    int up  = (int)lemire32(g, 1u);
    int ksv = ws[OFF_KS + q], dv = ws[OFF_DIL + q];
    int p = ((ksv - 1) * dv / 2) * up;
    ws[OFF_P2 + q] = 2 * p;
    ws[OFF_L  + q] = SEQ_LEN + 4 * p - dv * (ksv - 1);
  }
  // counting sort by (pad, ks, dil)
  for (int q = 0; q < N_KERNELS; ++q) ws[OFF_HIST + group_key(ws, q)]++;
  int run = 0;
  for (int k = 0; k < NKEYS; ++k) { int c = ws[OFF_HIST + k]; ws[OFF_HIST + k] = run; run += c; }
  for (int q = 0; q < N_KERNELS; ++q)
    ws[OFF_SORT + ws[OFF_HIST + group_key(ws, q)]++] = q;
  // tiles: runs of <=16 same-key kernels (all rows share dil, p2, L)
  int nt = 0, i = 0;
  while (i < N_KERNELS && nt < MAX_TILES) {
    int k0 = ws[OFF_SORT + i];
    int key0 = group_key(ws, k0);
    int j = i;
    while (j < N_KERNELS && (j - i) < 16 && group_key(ws, ws[OFF_SORT + j]) == key0) ++j;
    int* t = ws + OFF_TILE + 5 * nt;
    t[0] = i; t[1] = j - i; t[2] = ws[OFF_DIL + k0]; t[3] = ws[OFF_P2 + k0]; t[4] = ws[OFF_L + k0];
    ++nt; i = j;
  }
  ws[OFF_META] = nt;
}

// =========== main: WMMA f32 16x16x4  (M=time, K=taps, N=kernels) ===========
__global__ __launch_bounds__(32) void rocket_conv_wmma(
    const float* __restrict__ x, const float* __restrict__ W,
    const float* __restrict__ bias, float* __restrict__ out,
    const int* __restrict__ ws)
{
  const int numTiles = ws[OFF_META];
  const int tile = blockIdx.x;
  if (tile >= numTiles) return;                 // uniform exit (EXEC stays full)
  const int b = blockIdx.y;

  const int* tm = ws + OFF_TILE + 5 * tile;
  const int base = tm[0], rows = tm[1], dil = tm[2], p2 = tm[3], Lk = tm[4];

  const int lane = threadIdx.x;
  const int col = lane & 15;        // N (kernel) and A-row M index for this lane
  const int hh  = lane >> 4;        // lane-half selects K pair / M+8 in D

  __shared__ float xs[SEQ_LEN];
  const float* xrow = x + b * SEQ_LEN;
  for (int i = lane; i < SEQ_LEN; i += 32) xs[i] = xrow[i];

  const int r    = (col < rows) ? col : 0;      // pad short tiles with row 0
  const int kid  = ws[OFF_SORT + base + r];
  __builtin_prefetch(&W[kid * KMAX], 0, 1);     // global_prefetch_b8 on weight rows
  const float bv = bias[kid];

  // B (4x16): lanes0-15 hold K={0,1}, lanes16-31 K={2,3} per 4-tap chunk.
  // Taps >= ks are zero in W already; tap 11 padded to zero explicitly.
  v2f Bt[3];
#pragma unroll
  for (int jc = 0; jc < 3; ++jc) {
    int j0 = jc * 4 + hh * 2;
    Bt[jc].x = (j0     < KMAX) ? W[kid * KMAX + j0]     : 0.f;
    Bt[jc].y = (j0 + 1 < KMAX) ? W[kid * KMAX + j0 + 1] : 0.f;
  }
  __syncthreads();

  float kmax = -INFINITY, kcnt = 0.f;
  const int nfull = Lk >> 4;          // fully-valid 16-position chunks
  const int rem   = Lk & 15;

  // -------- full chunks: no validity masking needed (L uniform in tile) -----
  for (int c = 0; c < nfull; ++c) {
    const int tbase = c << 4;
    v8f acc = {0.f, 0.f, 0.f, 0.f, 0.f, 0.f, 0.f, 0.f};
#pragma unroll
    for (int jc = 0; jc < 3; ++jc) {
      int j0   = jc * 4 + hh * 2;
      int pos0 = tbase + col - p2 + j0 * dil;   // x index with zero-extension
      int pos1 = pos0 + dil;
      v2f a;
      a.x = ((unsigned)pos0 < (unsigned)SEQ_LEN) ? xs[pos0] : 0.f;
      a.y = ((unsigned)pos1 < (unsigned)SEQ_LEN) ? xs[pos1] : 0.f;
      acc = __builtin_amdgcn_wmma_f32_16x16x4_f32(
          false, a, false, Bt[jc], (short)0, acc, false, false);
    }
#pragma unroll
    for (int q = 0; q < 8; ++q) {               // lane<16: t=tbase+q ; lane>=16: +8
      float v = acc[q] + bv;
      kmax = fmaxf(kmax, v);
      kcnt += (v > 0.f) ? 1.f : 0.f;
    }
  }
  // -------- single masked tail chunk ----------------------------------------
  if (rem) {
    const int tbase = nfull << 4;
    v8f acc = {0.f, 0.f, 0.f, 0.f, 0.f, 0.f, 0.f, 0.f};
#pragma unroll
    for (int jc = 0; jc < 3; ++jc) {
      int j0   = jc * 4 + hh * 2;
      int pos0 = tbase + col - p2 + j0 * dil;
      int pos1 = pos0 + dil;
      v2f a;
      a.x = ((unsigned)pos0 < (unsigned)SEQ_LEN) ? xs[pos0] : 0.f;
      a.y = ((unsigned)pos1 < (unsigned)SEQ_LEN) ? xs[pos1] : 0.f;
      acc = __builtin_amdgcn_wmma_f32_16x16x4_f32(
          false, a, false, Bt[jc], (short)0, acc, false, false);
    }
    const int lim = rem - 8 * hh;               // valid q < lim for this lane-half
#pragma unroll
    for (int q = 0; q < 8; ++q) {
      float v = acc[q] + bv;
      bool valid = q < lim;
      kmax = (valid && v > kmax) ? v : kmax;
      kcnt += (valid && v > 0.f) ? 1.f : 0.f;
    }
  }
  // fold the two lane-halves (same kernel column, disjoint times)
  float om = __shfl_xor(kmax, 16, 32);
  kmax = (om > kmax) ? om : kmax;
  kcnt += __shfl_xor(kcnt, 16, 32);

  if (lane < rows) {
    out[b * (2 * N_KERNELS) + 2 * kid]     = kmax;
    out[b * (2 * N_KERNELS) + 2 * kid + 1] = kcnt / (float)Lk;
  }
}

// ================================ launch ================================
extern "C" void kernel_launch(void* const* d_in, const int* in_sizes, int n_in,
                              void* d_out, int out_size, void* d_ws, size_t ws_size,
                              hipStream_t stream) {
  (void)in_sizes; (void)n_in; (void)out_size; (void)ws_size;
  const float* x    = (const float*)d_in[0];
  const float* W    = (const float*)d_in[1];
  const float* bias = (const float*)d_in[2];
  float* out = (float*)d_out;
  int*   ws  = (int*)d_ws;

  rocket_init<<<1, 256, 0, stream>>>(ws);
  dim3 grid(MAX_TILES, BATCH);
  rocket_conv_wmma<<<grid, 32, 0, stream>>>(x, W, bias, out, ws);
}